// FCOSPostProcessor_57363583205468
// MI455X (gfx1250) — compile-verified
//
#include <hip/hip_runtime.h>
#include <stdint.h>

// ---------------- problem constants (fixed by the reference) ----------------
#define N_IMG   16
#define HH      256
#define WW      256
#define HW      (HH * WW)          // 65536 scores per image (C == 1)
#define K_TOP   1000
#define N_KPS   17
#define OUT_COLS 58                // 4 box + score + label + valid + 17*3 kps
#define NTHREADS 1024
#define ELEMS_PER_THREAD 64        // 65536 / 1024

// LDS layout: padded score rows (64 data words + 4 pad words) keep every
// b128 chunk 16B-aligned and make the strided per-thread reads only
// 2-way bank conflicted.
#define ROW_STRIDE 68
#define OFF_KEYS   0
#define OFF_SC     8192                               // 1024 * 8B keys
#define OFF_HIST   (OFF_SC + 1024 * ROW_STRIDE * 4)   // 278528 B of scores
#define OFF_SCAN   (OFF_HIST + 256 * 4)
#define OFF_VARS   (OFF_SCAN + NTHREADS * 4)
#define SMEM_BYTES (OFF_VARS + 16)

// CDNA5 async global->LDS path (device pass only; host pass takes fallback)
#if defined(__HIP_DEVICE_COMPILE__) && defined(__gfx1250__) && \
    __has_builtin(__builtin_amdgcn_global_load_async_to_lds_b128) && \
    __has_builtin(__builtin_amdgcn_s_wait_asynccnt)
#define USE_ASYNC_LDS 1
// builtin signature: (global v4i* src, local v4i* dst, imm offset, imm cpol)
typedef int v4i_t __attribute__((vector_size(16)));
typedef __attribute__((address_space(1))) v4i_t* gv4i_p;
typedef __attribute__((address_space(3))) v4i_t* lv4i_p;
#endif

__device__ __forceinline__ unsigned ldsWordOfElem(unsigned e) {
    return (e >> 6) * ROW_STRIDE + (e & 63u);
}
// monotonic float -> uint mapping (descending float == descending uint)
__device__ __forceinline__ unsigned orderKey(float f) {
    unsigned u = __float_as_uint(f);
    return (u & 0x80000000u) ? ~u : (u | 0x80000000u);
}
__device__ __forceinline__ float keyToFloat(unsigned ou) {
    unsigned u = (ou & 0x80000000u) ? (ou ^ 0x80000000u) : ~ou;
    return __uint_as_float(u);
}

extern "C" __global__ void __launch_bounds__(NTHREADS, 1)
fcos_topk_decode(const float* __restrict__ locations,
                 const float* __restrict__ box_cls,
                 const float* __restrict__ kps_pred,
                 const int*   __restrict__ image_sizes,
                 float*       __restrict__ out)
{
    extern __shared__ char smem[];
    uint64_t* keys = (uint64_t*)(smem + OFF_KEYS);
    unsigned* sc   = (unsigned*)(smem + OFF_SC);
    unsigned* hist = (unsigned*)(smem + OFF_HIST);
    unsigned* scan = (unsigned*)(smem + OFF_SCAN);
    unsigned* vars = (unsigned*)(smem + OFF_VARS); // [0]=prefix [1]=remK [2]=ctr

    const unsigned tid = threadIdx.x;
    const unsigned img = blockIdx.x;
    const float* plane = box_cls + (size_t)img * HW;

    // ---- Stage 1: stage the 256KB score plane into LDS ---------------------
#ifdef USE_ASYNC_LDS
    {
        unsigned ldsbase = (unsigned)(unsigned long long)(uintptr_t)smem + OFF_SC;
        unsigned long long gbase = (unsigned long long)(uintptr_t)plane;
        #pragma unroll
        for (int p = 0; p < 16; ++p) {
            unsigned c = (unsigned)p * NTHREADS + tid;     // b128 chunk (4 floats)
            unsigned ldsb = ldsbase + (c >> 4) * (ROW_STRIDE * 4u) + (c & 15u) * 16u;
            unsigned long long g = gbase + (unsigned long long)c * 16ull;
            __builtin_amdgcn_global_load_async_to_lds_b128(
                (gv4i_p)g, (lv4i_p)ldsb, 0, 0);
        }
        __builtin_amdgcn_s_wait_asynccnt(0);
    }
#else
    {
        const float4* gp = (const float4*)plane;
        #pragma unroll
        for (int p = 0; p < 16; ++p) {
            unsigned c = (unsigned)p * NTHREADS + tid;
            float4 v = gp[c];
            unsigned w0 = (c >> 4) * ROW_STRIDE + (c & 15u) * 4u;
            sc[w0 + 0] = __float_as_uint(v.x);
            sc[w0 + 1] = __float_as_uint(v.y);
            sc[w0 + 2] = __float_as_uint(v.z);
            sc[w0 + 3] = __float_as_uint(v.w);
        }
    }
#endif
    __syncthreads();

    // ---- Stage 2: in-place sigmoid + threshold-mask + order-key ------------
    #pragma unroll 4
    for (int i = 0; i < ELEMS_PER_THREAD; ++i) {
        unsigned w = ldsWordOfElem(tid * ELEMS_PER_THREAD + i);
        float x = __uint_as_float(sc[w]);
        float s = 1.0f / (1.0f + __expf(-x));
        float m = (s > 0.05f) ? s : -1.0f;
        sc[w] = orderKey(m);
    }
    if (tid == 0) { vars[0] = 0u; vars[1] = K_TOP; }
    __syncthreads();

    // ---- Stage 3: 4-level radix select for the exact K-th largest value ----
    for (int b = 3; b >= 0; --b) {
        if (tid < 256) hist[tid] = 0u;
        __syncthreads();
        const unsigned ph = vars[0];
        const unsigned shift = 8u * (unsigned)b;
        for (int i = 0; i < ELEMS_PER_THREAD; ++i) {
            unsigned ou = sc[ldsWordOfElem(tid * ELEMS_PER_THREAD + i)];
            unsigned hi = (b == 3) ? 0u : (ou >> (shift + 8u));
            if (hi == ph) atomicAdd(&hist[(ou >> shift) & 255u], 1u);
        }
        __syncthreads();
        if (tid == 0) {
            unsigned rem = vars[1], cum = 0u; int chosen = 0;
            for (int bin = 255; bin >= 0; --bin) {
                unsigned h = hist[bin];
                if (cum + h >= rem) { chosen = bin; break; }
                cum += h;
            }
            vars[0] = (ph << 8) | (unsigned)chosen;
            vars[1] = rem - cum;          // ties still needed at value T
        }
        __syncthreads();
    }
    const unsigned T = vars[0];
    const unsigned R = vars[1];

    // ---- Stage 4: compact winners into keys[] ------------------------------
    keys[tid] = 0ull;                     // pad slots sort last (all real keys > 0)
    if (tid == 0) vars[2] = 0u;
    __syncthreads();
    unsigned localEq = 0u;
    for (int i = 0; i < ELEMS_PER_THREAD; ++i) {
        unsigned e  = tid * ELEMS_PER_THREAD + (unsigned)i;
        unsigned ou = sc[ldsWordOfElem(e)];
        if (ou > T) {
            unsigned p = atomicAdd(&vars[2], 1u);
            keys[p] = ((uint64_t)ou << 32) | (uint64_t)(unsigned)(~e);
        } else if (ou == T) {
            localEq++;
        }
    }
    // exclusive scan of per-thread tie counts (contiguous ownership => scan
    // order == flattened index order, matching lax.top_k's stable ties)
    scan[tid] = localEq;
    __syncthreads();
    for (unsigned off = 1u; off < NTHREADS; off <<= 1) {
        unsigned add = (tid >= off) ? scan[tid - off] : 0u;
        __syncthreads();
        scan[tid] += add;
        __syncthreads();
    }
    unsigned rank = scan[tid] - localEq;
    for (int i = 0; i < ELEMS_PER_THREAD && rank < R; ++i) {
        unsigned e  = tid * ELEMS_PER_THREAD + (unsigned)i;
        unsigned ou = sc[ldsWordOfElem(e)];
        if (ou == T) {
            unsigned p = atomicAdd(&vars[2], 1u);
            keys[p] = ((uint64_t)T << 32) | (uint64_t)(unsigned)(~e);
            rank++;
        }
    }
    __syncthreads();

    // ---- Stage 5: bitonic sort 1024 keys descending ------------------------
    for (unsigned k = 2u; k <= NTHREADS; k <<= 1) {
        for (unsigned j = k >> 1; j > 0u; j >>= 1) {
            unsigned ixj = tid ^ j;
            if (ixj > tid) {
                uint64_t a = keys[tid], c = keys[ixj];
                bool desc = ((tid & k) == 0u);
                if (desc ? (a < c) : (a > c)) { keys[tid] = c; keys[ixj] = a; }
            }
            __syncthreads();
        }
    }

    // ---- Stage 6: decode + gather + write 58-float rows --------------------
    if (tid < K_TOP) {
        uint64_t key = keys[tid];
        unsigned ou  = (unsigned)(key >> 32);
        unsigned idx = ~(unsigned)(key & 0xFFFFFFFFull);
        float score  = keyToFloat(ou);
        float lx = locations[2u * idx + 0u];
        float ly = locations[2u * idx + 1u];
        const float* kp = kps_pred + (size_t)img * (size_t)(2 * N_KPS) * HW;
        float kx[N_KPS], ky[N_KPS];
        float minx = 1e30f, miny = 1e30f, maxx = -1e30f, maxy = -1e30f;
        #pragma unroll
        for (int k = 0; k < N_KPS; ++k) {
            float x = kp[(size_t)(2 * k + 0) * HW + idx] + lx;
            float y = kp[(size_t)(2 * k + 1) * HW + idx] + ly;
            kx[k] = x; ky[k] = y;
            minx = fminf(minx, x); maxx = fmaxf(maxx, x);
            miny = fminf(miny, y); maxy = fmaxf(maxy, y);
        }
        float hmax = (float)image_sizes[2u * img + 0u] - 1.0f;
        float wmax = (float)image_sizes[2u * img + 1u] - 1.0f;
        float x1 = fminf(fmaxf(minx, 0.0f), wmax);
        float y1 = fminf(fmaxf(miny, 0.0f), hmax);
        float x2 = fminf(fmaxf(maxx, 0.0f), wmax);
        float y2 = fminf(fmaxf(maxy, 0.0f), hmax);
        float ws = x2 - x1 + 1.0f, hs = y2 - y1 + 1.0f;
        float valid = (score > 0.05f && ws >= 0.0f && hs >= 0.0f) ? 1.0f : 0.0f;

        float* o = out + ((size_t)img * K_TOP + tid) * OUT_COLS;
        o[0] = x1; o[1] = y1; o[2] = x2; o[3] = y2;
        o[4] = score; o[5] = 1.0f; o[6] = valid;
        #pragma unroll
        for (int k = 0; k < N_KPS; ++k) {
            o[7 + 3 * k] = kx[k];
            o[8 + 3 * k] = ky[k];
            o[9 + 3 * k] = 1.0f;
        }
    }
}

extern "C" void kernel_launch(void* const* d_in, const int* in_sizes, int n_in,
                              void* d_out, int out_size, void* d_ws, size_t ws_size,
                              hipStream_t stream) {
    (void)in_sizes; (void)n_in; (void)out_size; (void)d_ws; (void)ws_size;
    const float* locations   = (const float*)d_in[0];
    const float* box_cls     = (const float*)d_in[1];
    // d_in[2] = centerness: sigmoid computed in reference but unused downstream
    const float* kps_pred    = (const float*)d_in[3];
    const int*   image_sizes = (const int*)d_in[4];
    float* out = (float*)d_out;

    // allow ~292KB dynamic LDS (CDNA5 WGP has 320KB); deterministic every call
    (void)hipFuncSetAttribute((const void*)fcos_topk_decode,
                              hipFuncAttributeMaxDynamicSharedMemorySize,
                              SMEM_BYTES);
    fcos_topk_decode<<<N_IMG, NTHREADS, SMEM_BYTES, stream>>>(
        locations, box_cls, kps_pred, image_sizes, out);
}